// FocalLoss_66949950210607
// MI455X (gfx1250) — compile-verified
//
#include <hip/hip_runtime.h>

typedef float v2f __attribute__((ext_vector_type(2)));
typedef float v8f __attribute__((ext_vector_type(8)));
typedef float f4  __attribute__((ext_vector_type(4)));

// Focal loss, coefficients folded out of the hot loop:
//   neg elem = -(1-a)*q^2*ln(1-q) = K_N * q^2*log2(1-q),   K_N = -0.75*ln2
//   pos elem = -a*(1-q)^2*ln(q)   = K_P * (1-q)^2*log2(q), K_P = -0.25*ln2
#define K_NEG (-0.5198603854199590f)   // -0.75 * ln(2)
#define K_POS (-0.1732867951399863f)   // -0.25 * ln(2)

// Wave32 sum reduction through the matrix unit: V_WMMA_F32_16X16X4_F32, B = ones.
// A (16x4): lane L<16 -> (M=L,K=0)=acc; lane L>=16 -> (M=L-16,K=2)=acc; K=1,3 zero.
// => D[m,n] = acc_m + acc_{m+16}. Sum of a lane's 8 D regs = half-wave total;
// xor-shuffle(16) completes it. Caller must be on a non-divergent path.
__device__ __forceinline__ float wave_sum_wmma(float acc) {
    v2f a; a.x = acc;  a.y = 0.0f;
    v2f b; b.x = 1.0f; b.y = 1.0f;
    v8f c = {};
    v8f d = __builtin_amdgcn_wmma_f32_16x16x4_f32(
        /*neg_a=*/false, a, /*neg_b=*/false, b,
        /*c_mod=*/(short)0, c, /*reuse_a=*/false, /*reuse_b=*/false);
    float s = ((d[0] + d[1]) + (d[2] + d[3])) + ((d[4] + d[5]) + (d[6] + d[7]));
    s += __shfl_xor(s, 16, 32);
    return s;
}

// Stage 1: stream whole rows. All elements accumulated as non-target
// (5 VALU/elem: v_max_num, v_sub, v_log, v_mul, v_fma); the lone target
// column per row is fixed up in a rarely-taken branch by its owning thread.
__global__ void __launch_bounds__(256)
focal_partial_kernel(const float* __restrict__ p,
                     const int*   __restrict__ classes,
                     float*       __restrict__ block_sums,
                     int B, int C) {
    const int tid = threadIdx.x;
    float accN = 0.0f;   // sum of q^2 * log2(1-q) over ALL elements
    float accP = 0.0f;   // sum of (1-q)^2 * log2(q) over target elements

    for (int row = blockIdx.x; row < B; row += gridDim.x) {
        const int cls = classes[row];                      // uniform -> scalar load
        const f4* rp4 = (const f4*)(p + (size_t)row * (size_t)C);
        for (int c4 = tid; c4 * 4 < C; c4 += 256) {
            f4 v = __builtin_nontemporal_load(&rp4[c4]);   // streaming: TH=NT
            // NaN -> 1e-6 in ONE op: maximumNumber(NaN, x) == x, and inputs
            // are >= 1e-4 so the clamp is otherwise inert.
            const float q0 = fmaxf(v.x, 1e-6f);
            const float q1 = fmaxf(v.y, 1e-6f);
            const float q2 = fmaxf(v.z, 1e-6f);
            const float q3 = fmaxf(v.w, 1e-6f);
            accN = fmaf(q0 * q0, __log2f(1.0f - q0), accN);
            accN = fmaf(q1 * q1, __log2f(1.0f - q1), accN);
            accN = fmaf(q2 * q2, __log2f(1.0f - q2), accN);
            accN = fmaf(q3 * q3, __log2f(1.0f - q3), accN);

            // Target-column fixup: taken by 1 of 512 float4 chunks per row.
            const int col = c4 * 4;
            const unsigned rel = (unsigned)(cls - col);
            if (rel < 4u) {
                const float qt = (rel == 0) ? q0 : (rel == 1) ? q1
                               : (rel == 2) ? q2 : q3;
                accN = fmaf(qt * qt, -__log2f(1.0f - qt), accN); // undo neg term
                const float omt = 1.0f - qt;
                accP = fmaf(omt * omt, __log2f(qt), accP);       // add pos term
            }
        }
    }

    // Apply folded coefficients once per thread, then reduce.
    const float acc = fmaf(accN, K_NEG, accP * K_POS);

    // Uniform path: every thread participates (EXEC all ones for WMMA).
    const float wsum = wave_sum_wmma(acc);

    __shared__ float smem[8];                              // 256 threads = 8 waves
    const int lane = tid & 31;
    const int wv   = tid >> 5;
    if (lane == 0) smem[wv] = wsum;
    __syncthreads();
    if (tid == 0) {
        float s = 0.0f;
        #pragma unroll
        for (int i = 0; i < 8; ++i) s += smem[i];
        block_sums[blockIdx.x] = s;
    }
}

// Stage 2: single block folds the per-block partials, scales by 1/B.
// Deterministic (fixed order, no float atomics).
__global__ void __launch_bounds__(256)
focal_final_kernel(const float* __restrict__ block_sums,
                   float*       __restrict__ out,
                   int n, float scale) {
    const int tid = threadIdx.x;
    float acc = 0.0f;
    for (int i = tid; i < n; i += 256) acc += block_sums[i];

    const float wsum = wave_sum_wmma(acc);                 // uniform path

    __shared__ float smem[8];
    const int lane = tid & 31;
    const int wv   = tid >> 5;
    if (lane == 0) smem[wv] = wsum;
    __syncthreads();
    if (tid == 0) {
        float s = 0.0f;
        #pragma unroll
        for (int i = 0; i < 8; ++i) s += smem[i];
        out[0] = s * scale;
    }
}

extern "C" void kernel_launch(void* const* d_in, const int* in_sizes, int n_in,
                              void* d_out, int out_size, void* d_ws, size_t ws_size,
                              hipStream_t stream) {
    const float* p       = (const float*)d_in[0];
    const int*   classes = (const int*)d_in[1];
    const int total = in_sizes[0];          // B*C
    const int B     = in_sizes[1];
    const int C     = total / B;            // 2048

    float* ws  = (float*)d_ws;
    float* out = (float*)d_out;

    int nb = B < 2048 ? B : 2048;           // 16 rows per block at B=32768

    focal_partial_kernel<<<dim3(nb), dim3(256), 0, stream>>>(p, classes, ws, B, C);
    focal_final_kernel<<<dim3(1), dim3(256), 0, stream>>>(ws, out, nb, 1.0f / (float)B);
}